// DnnFull_72026601554182
// MI455X (gfx1250) — compile-verified
//
#include <hip/hip_runtime.h>
#include <math.h>

// ---------------------------------------------------------------------------
// Problem constants (match reference setup_inputs exactly)
// ---------------------------------------------------------------------------
constexpr int N_ATOMS = 64;
constexpr int T_TYPES = 2;
constexpr int M_COLS  = 27 * N_ATOMS;                 // 1728
constexpr int P_PAIR  = T_TYPES * (T_TYPES + 1) / 2;  // 3
constexpr float CUT   = 25.0f;
constexpr float SLOPE = 0.01f;                        // jax.nn.leaky_relu default

// ED = [3, 32, 64, 128], LD = [384, 256, 256, 128, 3]
constexpr int E0 = 3, E1 = 32, E2 = 64, E3 = 128;
constexpr int D_FEAT = E3 * 3;                        // 384

// Packed-B fragment counts (v2f per (p,kc,ot,lane))
constexpr int BP1_V2 = P_PAIR * 1 * (E1 / 16) * 32;            // 192
constexpr int BP2_V2 = P_PAIR * (E1 / 4) * (E2 / 16) * 32;     // 3072
constexpr int BP3_V2 = P_PAIR * (E2 / 4) * (E3 / 16) * 32;     // 12288

// Workspace layout (in floats; all offsets even -> 8B aligned for v2f)
constexpr int OFF_XEXT = 0;                               // [M][3]
constexpr int OFF_KEEP = OFF_XEXT + M_COLS * 3;           // [N][M]
constexpr int OFF_ORD  = OFF_KEEP + N_ATOMS * M_COLS;     // int [M]
constexpr int OFF_GOFF = OFF_ORD + M_COLS;                // int [4]
constexpr int OFF_DACC = OFF_GOFF + 4;                    // [N][384]
constexpr int OFF_OUTP = OFF_DACC + N_ATOMS * D_FEAT;     // [N][3]
constexpr int OFF_BP1  = OFF_OUTP + N_ATOMS * 3;          // 2*BP1_V2
constexpr int OFF_BP2  = OFF_BP1 + 2 * BP1_V2;
constexpr int OFF_BP3  = OFF_BP2 + 2 * BP2_V2;
// end: OFF_BP3 + 2*BP3_V2

typedef float v2f __attribute__((ext_vector_type(2)));
typedef float v8f __attribute__((ext_vector_type(8)));

__device__ __forceinline__ v8f wmma_f32(v2f a, v2f b, v8f c) {
  // D = A(16x4,f32) * B(4x16,f32) + C(16x16,f32)
  return __builtin_amdgcn_wmma_f32_16x16x4_f32(false, a, false, b, (short)0, c,
                                               false, false);
}

__device__ __forceinline__ float lrelu(float x) {
  return x >= 0.0f ? x : SLOPE * x;
}

// ---------------------------------------------------------------------------
// prep1: zero the d-accumulator and build x_ext = x + combos @ lattice
// ---------------------------------------------------------------------------
__global__ void prep1_kernel(const float* __restrict__ x,
                             const float* __restrict__ lattice,
                             float* __restrict__ x_ext,
                             float* __restrict__ dacc) {
  int tid = blockIdx.x * blockDim.x + threadIdx.x;
  if (tid < N_ATOMS * D_FEAT) dacc[tid] = 0.0f;
  if (tid < M_COLS) {
    int k = tid / N_ATOMS;  // combo index (combo-major, matches reshape)
    int i = tid % N_ATOMS;  // atom index
    float ca = (float)(k / 9) - 1.0f;
    float cb = (float)((k / 3) % 3) - 1.0f;
    float cc = (float)(k % 3) - 1.0f;
    #pragma unroll
    for (int c = 0; c < 3; ++c) {
      float off = ca * lattice[0 * 3 + c] + cb * lattice[1 * 3 + c] +
                  cc * lattice[2 * 3 + c];
      x_ext[tid * 3 + c] = x[i * 3 + c] + off;
    }
  }
}

// ---------------------------------------------------------------------------
// pack: pre-swizzle We1/We2/We3 into WMMA B-fragment order.
// Fragment layout: bpL[((p*KC + kc)*OT + ot)*32 + lane] = v2f{W[k0][col],
// W[k0+1][col]} with k0 = kc*4 + 2*(lane>>4), col = ot*16 + (lane&15).
// Layer-1 K padding (K=3 -> 4) is baked in as zeros.
// ---------------------------------------------------------------------------
__global__ void pack_kernel(const float* __restrict__ We1,
                            const float* __restrict__ We2,
                            const float* __restrict__ We3,
                            float* __restrict__ bp1,
                            float* __restrict__ bp2,
                            float* __restrict__ bp3) {
  int tid = blockIdx.x * blockDim.x + threadIdx.x;
  if (tid < BP1_V2) {  // P * 2ot * 32
    int lane = tid & 31;
    int ot = (tid >> 5) & 1;
    int p = tid >> 6;
    int lo = lane & 15, hi = lane >> 4;
    int k0 = 2 * hi, col = ot * 16 + lo;
    bp1[tid * 2 + 0] = (k0 < E0) ? We1[(p * E0 + k0) * E1 + col] : 0.0f;
    bp1[tid * 2 + 1] = (k0 + 1 < E0) ? We1[(p * E0 + k0 + 1) * E1 + col] : 0.0f;
  }
  if (tid < BP2_V2) {  // P * 8kc * 4ot * 32
    int lane = tid & 31;
    int ot = (tid >> 5) & 3;
    int kc = (tid >> 7) & 7;
    int p = tid >> 10;
    int lo = lane & 15, hi = lane >> 4;
    int k0 = kc * 4 + 2 * hi, col = ot * 16 + lo;
    bp2[tid * 2 + 0] = We2[(p * E1 + k0) * E2 + col];
    bp2[tid * 2 + 1] = We2[(p * E1 + k0 + 1) * E2 + col];
  }
  if (tid < BP3_V2) {  // P * 16kc * 8ot * 32
    int lane = tid & 31;
    int ot = (tid >> 5) & 7;
    int kc = (tid >> 8) & 15;
    int p = tid >> 12;
    int lo = lane & 15, hi = lane >> 4;
    int k0 = kc * 4 + 2 * hi, col = ot * 16 + lo;
    bp3[tid * 2 + 0] = We3[(p * E2 + k0) * E3 + col];
    bp3[tid * 2 + 1] = We3[(p * E2 + k0 + 1) * E3 + col];
  }
}

// ---------------------------------------------------------------------------
// prep2: keep-mask (with the reference's cumsum/rank quirk) + column sort by
// type (ord, group offsets). One thread per central atom; thread 0 sorts.
// ---------------------------------------------------------------------------
__global__ void prep2_kernel(const float* __restrict__ x_ext,
                             const int* __restrict__ atom_types,
                             float* __restrict__ keepB,
                             int* __restrict__ ord,
                             int* __restrict__ goff) {
  int n = threadIdx.x;
  if (n == 0) {
    int cnt[T_TYPES];
    for (int t = 0; t < T_TYPES; ++t) cnt[t] = 0;
    for (int m = 0; m < M_COLS; ++m) cnt[atom_types[m % N_ATOMS]]++;
    goff[0] = 0;
    for (int t = 0; t < T_TYPES; ++t) goff[t + 1] = goff[t] + cnt[t];
    int pos[T_TYPES];
    for (int t = 0; t < T_TYPES; ++t) pos[t] = goff[t];
    for (int m = 0; m < M_COLS; ++m) ord[pos[atom_types[m % N_ATOMS]]++] = m;
  }
  if (n < N_ATOMS) {
    float xn0 = x_ext[n * 3 + 0];
    float xn1 = x_ext[n * 3 + 1];
    float xn2 = x_ext[n * 3 + 2];
    int cum = 0;
    for (int m = 0; m < M_COLS; ++m) {
      float dx = x_ext[m * 3 + 0] - xn0;
      float dy = x_ext[m * 3 + 1] - xn1;
      float dz = x_ext[m * 3 + 2] - xn2;
      float dist = sqrtf(dx * dx + dy * dy + dz * dz);
      bool mask = dist <= CUT;
      cum += mask ? 1 : 0;
      int rank = cum - 1;
      keepB[n * M_COLS + m] = (mask && rank != n) ? 1.0f : 0.0f;
    }
  }
}

// ---------------------------------------------------------------------------
// embed: one wave (32 threads) per 16-row tile of (atom n, column-type group).
// 3 GEMM layers on v_wmma_f32_16x16x4_f32 with pre-swizzled B fragments
// (one coalesced b64 load per WMMA), then the keep*g*rel contraction reduced
// in-wave and atomically accumulated into dacc[n][384].
// ---------------------------------------------------------------------------
__global__ __launch_bounds__(32) void embed_kernel(
    const float* __restrict__ x_ext, const float* __restrict__ keepB,
    const int* __restrict__ ord, const int* __restrict__ goff,
    const v2f* __restrict__ Bp1, const float* __restrict__ be1,
    const v2f* __restrict__ Bp2, const float* __restrict__ be2,
    const v2f* __restrict__ Bp3, const float* __restrict__ be3,
    const int* __restrict__ atom_types, float* __restrict__ dacc) {
  const int tile = blockIdx.x;
  const int tg   = blockIdx.y;  // column type group
  const int n    = blockIdx.z;

  const int r0   = goff[tg] + tile * 16;
  const int rend = goff[tg + 1];
  if (r0 >= rend) return;

  const int lane = threadIdx.x;
  const int lo = lane & 15;  // N-column / M-row selector
  const int hi = lane >> 4;  // half-wave selector

  // pair id from (type[n], group type tg)
  const int tn = atom_types[n];
  const int tl = tn < tg ? tn : tg;
  const int th = tn < tg ? tg : tn;
  const int p  = tl * T_TYPES - tl * (tl - 1) / 2 + (th - tl);

  // Fragment bases for this pair type
  const v2f* bp1 = Bp1 + (size_t)p * (E1 / 16) * 32 + lane;
  const v2f* bp2 = Bp2 + (size_t)p * (E1 / 4) * (E2 / 16) * 32 + lane;
  const v2f* bp3 = Bp3 + (size_t)p * (E2 / 4) * (E3 / 16) * 32 + lane;
  __builtin_prefetch(bp3, 0, 0);
  __builtin_prefetch(bp2, 0, 0);

  __shared__ float lds_h[16][E2];  // activation staging (16 x 64)
  __shared__ float lds_rel[16][3];
  __shared__ float lds_keep[16];

  // ---- gather the 16 rows of this tile -----------------------------------
  if (lane < 16) {
    int r = r0 + lane;
    float rv0 = 0.f, rv1 = 0.f, rv2 = 0.f, kp = 0.f;
    if (r < rend) {
      int mm = ord[r];
      rv0 = x_ext[mm * 3 + 0] - x_ext[n * 3 + 0];
      rv1 = x_ext[mm * 3 + 1] - x_ext[n * 3 + 1];
      rv2 = x_ext[mm * 3 + 2] - x_ext[n * 3 + 2];
      kp  = keepB[n * M_COLS + mm];
    }
    lds_rel[lane][0] = rv0;
    lds_rel[lane][1] = rv1;
    lds_rel[lane][2] = rv2;
    lds_keep[lane] = kp;
  }
  __syncthreads();

  // ---- layer 1: (16x3) x (3x32), K padded to 4 ---------------------------
  {
    const int k0 = 2 * hi;  // lanes 0-15 hold K=0,1; lanes 16-31 hold K=2,3
    v2f a;
    a.x = lds_rel[lo][k0];
    a.y = (k0 + 1 < 3) ? lds_rel[lo][k0 + 1] : 0.0f;
    #pragma unroll
    for (int ot = 0; ot < E1 / 16; ++ot) {
      v2f b = bp1[ot * 32];
      v8f c = {};
      c = wmma_f32(a, b, c);
      int col = ot * 16 + lo;
      float bias = be1[p * E1 + col];
      #pragma unroll
      for (int v = 0; v < 8; ++v)
        lds_h[v + 8 * hi][col] = lrelu(c[v] + bias);
    }
  }
  __syncthreads();

  // ---- layer 2: (16x32) x (32x64) ----------------------------------------
  v8f c2[E2 / 16];
  #pragma unroll
  for (int ot = 0; ot < E2 / 16; ++ot) {
    v8f c = {};
    #pragma unroll
    for (int kc = 0; kc < E1 / 4; ++kc) {
      int k0 = kc * 4 + 2 * hi;
      v2f a = {lds_h[lo][k0], lds_h[lo][k0 + 1]};
      v2f b = bp2[(kc * (E2 / 16) + ot) * 32];
      c = wmma_f32(a, b, c);
    }
    int col = ot * 16 + lo;
    float bias = be2[p * E2 + col];
    #pragma unroll
    for (int v = 0; v < 8; ++v) c[v] = lrelu(c[v] + bias);
    c2[ot] = c;
  }
  __syncthreads();
  #pragma unroll
  for (int ot = 0; ot < E2 / 16; ++ot) {
    int col = ot * 16 + lo;
    #pragma unroll
    for (int v = 0; v < 8; ++v) lds_h[v + 8 * hi][col] = c2[ot][v];
  }
  __syncthreads();

  // ---- layer 3: (16x64) x (64x128) ---------------------------------------
  v8f c3[E3 / 16];
  #pragma unroll
  for (int ot = 0; ot < E3 / 16; ++ot) {
    v8f c = {};
    #pragma unroll
    for (int kc = 0; kc < E2 / 4; ++kc) {
      int k0 = kc * 4 + 2 * hi;
      v2f a = {lds_h[lo][k0], lds_h[lo][k0 + 1]};
      v2f b = bp3[(kc * (E3 / 16) + ot) * 32];
      c = wmma_f32(a, b, c);
    }
    int col = ot * 16 + lo;
    float bias = be3[p * E3 + col];
    #pragma unroll
    for (int v = 0; v < 8; ++v) c[v] = lrelu(c[v] + bias);
    c3[ot] = c;
  }

  // ---- contraction: d[n][e][c] += sum_rows keep * g * rel ----------------
  #pragma unroll
  for (int ot = 0; ot < E3 / 16; ++ot) {
    float pc0 = 0.f, pc1 = 0.f, pc2 = 0.f;
    #pragma unroll
    for (int v = 0; v < 8; ++v) {
      int row = v + 8 * hi;
      float gk = c3[ot][v] * lds_keep[row];
      pc0 += gk * lds_rel[row][0];
      pc1 += gk * lds_rel[row][1];
      pc2 += gk * lds_rel[row][2];
    }
    pc0 += __shfl_xor(pc0, 16, 32);
    pc1 += __shfl_xor(pc1, 16, 32);
    pc2 += __shfl_xor(pc2, 16, 32);
    if (hi == 0) {
      int e = ot * 16 + lo;
      atomicAdd(&dacc[(size_t)n * D_FEAT + e * 3 + 0], pc0);
      atomicAdd(&dacc[(size_t)n * D_FEAT + e * 3 + 1], pc1);
      atomicAdd(&dacc[(size_t)n * D_FEAT + e * 3 + 2], pc2);
    }
  }
}

// ---------------------------------------------------------------------------
// fit: per-atom MLP 384 -> 256 -> 256 -> 128 -> 3 with per-atom weights.
// One 256-thread block per atom (GEMV chains; weights differ per atom so
// WMMA cannot batch across atoms). ~25 MFLOP total, negligible.
// ---------------------------------------------------------------------------
__global__ __launch_bounds__(256) void fit_kernel(
    const float* __restrict__ dacc,
    const float* __restrict__ Wl1, const float* __restrict__ bl1,
    const float* __restrict__ Wl2, const float* __restrict__ bl2,
    const float* __restrict__ Wl3, const float* __restrict__ bl3,
    const float* __restrict__ Wl4, const float* __restrict__ bl4,
    float* __restrict__ outp) {
  const int n = blockIdx.x;
  const int t = threadIdx.x;
  __shared__ float sIn[D_FEAT];  // 384
  __shared__ float sMid[256];

  for (int j = t; j < D_FEAT; j += 256) sIn[j] = dacc[(size_t)n * D_FEAT + j];
  __syncthreads();

  // layer 1: 384 -> 256
  {
    float acc = bl1[n * 256 + t];
    const float* W = Wl1 + (size_t)n * D_FEAT * 256;
    for (int i = 0; i < D_FEAT; ++i) acc += sIn[i] * W[(size_t)i * 256 + t];
    sMid[t] = lrelu(acc);
  }
  __syncthreads();
  // layer 2: 256 -> 256
  {
    float acc = bl2[n * 256 + t];
    const float* W = Wl2 + (size_t)n * 256 * 256;
    for (int i = 0; i < 256; ++i) acc += sMid[i] * W[(size_t)i * 256 + t];
    __syncthreads();
    sIn[t] = lrelu(acc);
  }
  __syncthreads();
  // layer 3: 256 -> 128
  if (t < 128) {
    float acc = bl3[n * 128 + t];
    const float* W = Wl3 + (size_t)n * 256 * 128;
    for (int i = 0; i < 256; ++i) acc += sIn[i] * W[(size_t)i * 128 + t];
    sMid[t] = lrelu(acc);
  }
  __syncthreads();
  // layer 4: 128 -> 3 (no activation)
  if (t < 3) {
    float acc = bl4[n * 3 + t];
    const float* W = Wl4 + (size_t)n * 128 * 3;
    for (int i = 0; i < 128; ++i) acc += sMid[i] * W[(size_t)i * 3 + t];
    outp[n * 3 + t] = acc;
  }
}

// ---------------------------------------------------------------------------
// mean-subtract over atoms, write final flattened output (192 floats)
// ---------------------------------------------------------------------------
__global__ void mean_kernel(const float* __restrict__ outp,
                            float* __restrict__ out) {
  __shared__ float mu[3];
  int t = threadIdx.x;
  if (t < 3) {
    float s = 0.f;
    for (int n = 0; n < N_ATOMS; ++n) s += outp[n * 3 + t];
    mu[t] = s / (float)N_ATOMS;
  }
  __syncthreads();
  if (t < N_ATOMS * 3) out[t] = outp[t] - mu[t % 3];
}

// ---------------------------------------------------------------------------
extern "C" void kernel_launch(void* const* d_in, const int* in_sizes, int n_in,
                              void* d_out, int out_size, void* d_ws,
                              size_t ws_size, hipStream_t stream) {
  const float* x       = (const float*)d_in[0];
  const float* lattice = (const float*)d_in[1];
  const float* We1 = (const float*)d_in[2];
  const float* be1 = (const float*)d_in[3];
  const float* We2 = (const float*)d_in[4];
  const float* be2 = (const float*)d_in[5];
  const float* We3 = (const float*)d_in[6];
  const float* be3 = (const float*)d_in[7];
  const float* Wl1 = (const float*)d_in[8];
  const float* bl1 = (const float*)d_in[9];
  const float* Wl2 = (const float*)d_in[10];
  const float* bl2 = (const float*)d_in[11];
  const float* Wl3 = (const float*)d_in[12];
  const float* bl3 = (const float*)d_in[13];
  const float* Wl4 = (const float*)d_in[14];
  const float* bl4 = (const float*)d_in[15];
  const int* atom_types = (const int*)d_in[16];

  float* ws    = (float*)d_ws;
  float* x_ext = ws + OFF_XEXT;
  float* keepB = ws + OFF_KEEP;
  int*   ord   = (int*)(ws + OFF_ORD);
  int*   goff  = (int*)(ws + OFF_GOFF);
  float* dacc  = ws + OFF_DACC;
  float* outp  = ws + OFF_OUTP;
  float* bp1   = ws + OFF_BP1;
  float* bp2   = ws + OFF_BP2;
  float* bp3   = ws + OFF_BP3;

  // 1) x_ext + zero accumulator; pre-swizzle embedding weights into B frags
  prep1_kernel<<<dim3((N_ATOMS * D_FEAT + 255) / 256), dim3(256), 0, stream>>>(
      x, lattice, x_ext, dacc);
  pack_kernel<<<dim3((BP3_V2 + 255) / 256), dim3(256), 0, stream>>>(
      We1, We2, We3, bp1, bp2, bp3);
  // 2) keep mask + type-sorted column order
  prep2_kernel<<<dim3(1), dim3(N_ATOMS), 0, stream>>>(x_ext, atom_types, keepB,
                                                      ord, goff);
  // 3) embedding net on WMMA: one wave per 16-row tile; worst-case tile count
  const int max_tiles = (M_COLS + 15) / 16;  // 108
  embed_kernel<<<dim3(max_tiles, T_TYPES, N_ATOMS), dim3(32), 0, stream>>>(
      x_ext, keepB, ord, goff, (const v2f*)bp1, be1, (const v2f*)bp2, be2,
      (const v2f*)bp3, be3, atom_types, dacc);
  // 4) per-atom fitting net
  fit_kernel<<<dim3(N_ATOMS), dim3(256), 0, stream>>>(
      dacc, Wl1, bl1, Wl2, bl2, Wl3, bl3, Wl4, bl4, outp);
  // 5) mean subtraction -> final output
  mean_kernel<<<dim3(1), dim3(192), 0, stream>>>(outp, (float*)d_out);
}